// baseIGNN_31044023616073
// MI455X (gfx1250) — compile-verified
//
#include <hip/hip_runtime.h>
#include <hip/hip_bf16.h>

// CDNA5 / gfx1250, wave32. FP32 WMMA (V_WMMA_F32_16X16X4_F32) for both GEMMs;
// scatter phase is wave-per-edge float4 gather + global_atomic_add_f32.

typedef float v2f __attribute__((ext_vector_type(2)));
typedef float v8f __attribute__((ext_vector_type(8)));

#define IN_CH  128
#define OUT_CH 128

// ---------------------------------------------------------------------------
// Kernel 1: deg[i] = 1.0 (self-loop seed)
// ---------------------------------------------------------------------------
__global__ void init_deg_kernel(float* __restrict__ deg, int n) {
    int i = blockIdx.x * blockDim.x + threadIdx.x;
    if (i < n) deg[i] = 1.0f;
}

// ---------------------------------------------------------------------------
// Kernel 2: deg[col[e]] += 1 for every edge
// ---------------------------------------------------------------------------
__global__ void count_deg_kernel(const int* __restrict__ col, float* __restrict__ deg, int e) {
    int i = blockIdx.x * blockDim.x + threadIdx.x;
    if (i < e) atomicAdd(&deg[col[i]], 1.0f);
}

// ---------------------------------------------------------------------------
// Kernel 3: dinv[i] = rsqrt(deg[i])   (deg >= 1 always, so no clamp needed)
// ---------------------------------------------------------------------------
__global__ void dinv_kernel(float* __restrict__ deg, int n) {
    int i = blockIdx.x * blockDim.x + threadIdx.x;
    if (i < n) deg[i] = __frsqrt_rn(deg[i]);
}

// ---------------------------------------------------------------------------
// Kernel 4: xs = (embedding @ conv_w^T) * dinv[row]; acc = xs (self-loop seed)
// One wave per 16x16 output tile; 8 waves/block cover the 128-wide output.
// A-frag (16x4 f32):  lane l<16 -> M=l, v0=K0,v1=K1 ; lane>=16 -> v0=K2,v1=K3
// B-frag (4x16 f32):  mirrored; B[k][n] = W[n][k] (row-major W -> contiguous pair)
// C/D    (16x16 f32): VGPR r: lanes0-15 M=r, lanes16-31 M=r+8, N=lane&15
// ---------------------------------------------------------------------------
__global__ __launch_bounds__(256) void gemm_xs_kernel(
        const float* __restrict__ emb,   // [N,128]
        const float* __restrict__ W,     // [128,128] row-major (conv_w)
        const float* __restrict__ dinv,  // [N]
        float* __restrict__ xs,          // [N,128]
        float* __restrict__ acc) {       // [N,128]
    const int lane = threadIdx.x & 31;
    const int wave = threadIdx.x >> 5;       // 0..7 -> column tile
    const int row0 = blockIdx.x * 16;
    const int col0 = wave * 16;
    const int half = lane >> 4;              // 0 or 1
    const int l    = lane & 15;

    const float* Abase = emb + (size_t)(row0 + l) * IN_CH + 2 * half;
    const float* Bbase = W   + (size_t)(col0 + l) * IN_CH + 2 * half;

    v8f c = {};
    #pragma unroll
    for (int s = 0; s < IN_CH / 4; ++s) {
        const int k = 4 * s;
        v2f a, b;
        a.x = Abase[k]; a.y = Abase[k + 1];
        b.x = Bbase[k]; b.y = Bbase[k + 1];
        c = __builtin_amdgcn_wmma_f32_16x16x4_f32(
                /*neg_a=*/false, a, /*neg_b=*/false, b,
                /*c_mod=*/(short)0, c, /*reuse_a=*/false, /*reuse_b=*/false);
    }

    #pragma unroll
    for (int r = 0; r < 8; ++r) {
        const int m = row0 + r + 8 * half;
        const float v = c[r] * dinv[m];
        const size_t o = (size_t)m * OUT_CH + col0 + l;
        xs[o]  = v;
        acc[o] = v;   // self-loop contribution (x[i]*dinv[i]); dinv[col] applied later
    }
}

// ---------------------------------------------------------------------------
// Kernel 5: edge scatter. One wave32 per edge: 32 lanes x float4 = 512B row.
//   acc[col] += xs[row]   (norm dinv[row] is pre-folded; dinv[col] folded later)
// ---------------------------------------------------------------------------
__global__ void scatter_kernel(const int* __restrict__ rows,
                               const int* __restrict__ cols,
                               const float* __restrict__ xs,
                               float* __restrict__ acc, int e) {
    const int gid  = blockIdx.x * blockDim.x + threadIdx.x;
    const int edge = gid >> 5;
    const int lane = gid & 31;
    if (edge >= e) return;
    const int r = rows[edge];
    const int cdst = cols[edge];
    const float4 v = *(const float4*)(xs + (size_t)r * OUT_CH + lane * 4);
    float* dst = acc + (size_t)cdst * OUT_CH + lane * 4;
    atomicAdd(dst + 0, v.x);
    atomicAdd(dst + 1, v.y);
    atomicAdd(dst + 2, v.z);
    atomicAdd(dst + 3, v.w);
}

// ---------------------------------------------------------------------------
// Kernel 6: out = relu( acc*dinv[row] + feature @ mlp_w^T )
// Same WMMA tiling as kernel 4; accumulator seeded with acc*dinv.
// ---------------------------------------------------------------------------
__global__ __launch_bounds__(256) void gemm_out_kernel(
        const float* __restrict__ feat,  // [N,128]
        const float* __restrict__ W,     // [128,128] row-major (mlp_w)
        const float* __restrict__ acc,   // [N,128]
        const float* __restrict__ dinv,  // [N]
        float* __restrict__ out) {       // [N,128]
    const int lane = threadIdx.x & 31;
    const int wave = threadIdx.x >> 5;
    const int row0 = blockIdx.x * 16;
    const int col0 = wave * 16;
    const int half = lane >> 4;
    const int l    = lane & 15;

    const float* Abase = feat + (size_t)(row0 + l) * IN_CH + 2 * half;
    const float* Bbase = W    + (size_t)(col0 + l) * IN_CH + 2 * half;

    // Seed C with the normalized aggregation term.
    v8f c;
    #pragma unroll
    for (int r = 0; r < 8; ++r) {
        const int m = row0 + r + 8 * half;
        c[r] = acc[(size_t)m * OUT_CH + col0 + l] * dinv[m];
    }

    #pragma unroll
    for (int s = 0; s < IN_CH / 4; ++s) {
        const int k = 4 * s;
        v2f a, b;
        a.x = Abase[k]; a.y = Abase[k + 1];
        b.x = Bbase[k]; b.y = Bbase[k + 1];
        c = __builtin_amdgcn_wmma_f32_16x16x4_f32(
                false, a, false, b, (short)0, c, false, false);
    }

    #pragma unroll
    for (int r = 0; r < 8; ++r) {
        const int m = row0 + r + 8 * half;
        out[(size_t)m * OUT_CH + col0 + l] = fmaxf(c[r], 0.0f);
    }
}

// ---------------------------------------------------------------------------
extern "C" void kernel_launch(void* const* d_in, const int* in_sizes, int n_in,
                              void* d_out, int out_size, void* d_ws, size_t ws_size,
                              hipStream_t stream) {
    const float* feature   = (const float*)d_in[0];          // [N,128]
    const int*   edge_idx  = (const int*)d_in[1];            // [2,E] int32
    const float* embedding = (const float*)d_in[2];          // [N,128]
    const float* conv_w    = (const float*)d_in[3];          // [128,128]
    const float* mlp_w     = (const float*)d_in[4];          // [128,128]
    float*       out       = (float*)d_out;

    const int N = in_sizes[0] / IN_CH;                       // 100000
    const int E = in_sizes[1] / 2;                           // 1600000
    const int* rows = edge_idx;          // edge_index[0] = source
    const int* cols = edge_idx + E;      // edge_index[1] = target

    // Workspace layout (floats): xs[N*128] | acc[N*128] | deg/dinv[N]
    float* xs   = (float*)d_ws;
    float* acc  = xs  + (size_t)N * OUT_CH;
    float* dinv = acc + (size_t)N * OUT_CH;

    // 1) degrees (self-loop seed = 1)
    init_deg_kernel<<<(N + 255) / 256, 256, 0, stream>>>(dinv, N);
    // 2) count in-degree per target node
    count_deg_kernel<<<(E + 255) / 256, 256, 0, stream>>>(cols, dinv, E);
    // 3) dinv = rsqrt(deg)  (in place)
    dinv_kernel<<<(N + 255) / 256, 256, 0, stream>>>(dinv, N);
    // 4) xs = (embedding @ conv_w^T)*dinv ; acc = xs  (WMMA, N multiple of 16)
    gemm_xs_kernel<<<N / 16, 256, 0, stream>>>(embedding, conv_w, dinv, xs, acc);
    // 5) edge scatter: acc[col] += xs[row]   (wave32 per edge)
    {
        const long long total = (long long)E * 32;
        scatter_kernel<<<(int)((total + 255) / 256), 256, 0, stream>>>(rows, cols, xs, acc, E);
    }
    // 6) out = relu(acc*dinv + feature @ mlp_w^T)  (WMMA)
    gemm_out_kernel<<<N / 16, 256, 0, stream>>>(feature, mlp_w, acc, dinv, out);
}